// Self_Attention_Local_72859825209374
// MI455X (gfx1250) — compile-verified
//
#include <hip/hip_runtime.h>

// ---------------------------------------------------------------------------
// MI455X / gfx1250, wave32. Matrix ops via V_WMMA_F32_16X16X4_F32 (fp32 in/out,
// matches the fp32 reference exactly up to accumulation order).
//
// Problem sizes (fixed by the reference):
//   patches NP=512, tokens/patch P=8, channels C=128, heads H=8, CH=1024
//   emb (512,8,128)  Wq/Wk/Wv (128,1024)  Wout (1024,128)  out (16,256,128)
//
// Workspace layout (fp32, 64 MB total):
//   [0 .. 4M)    Qw  [patch][head][p][c]
//   [4M .. 8M)   Kw  [patch][head][p][c]
//   [8M .. 12M)  Vw  [patch][head][p][c]
//   [12M .. 16M) Ctx [patch][p][c*8+h]   (rows match out-proj GEMM M dim)
// ---------------------------------------------------------------------------

typedef float v2f __attribute__((ext_vector_type(2)));
typedef float v8f __attribute__((ext_vector_type(8)));

#define WMMA_F32(A, B, Cacc) \
    __builtin_amdgcn_wmma_f32_16x16x4_f32(false, (A), false, (B), (short)0, (Cacc), false, false)

static constexpr int NP     = 512;              // patches
static constexpr int PTOK   = 8;                // tokens per patch
static constexpr int CCH    = 128;              // channels c
static constexpr int HEADS  = 8;
static constexpr int CH     = 1024;             // c*heads
static constexpr int MROWS  = NP * PTOK;        // 4096
static constexpr size_t SEG = (size_t)NP * HEADS * PTOK * CCH;  // 4,194,304 floats

// ---------------------------------------------------------------------------
// Kernel 1: QKV projection.  One wave per 16x16 tile of (4096 x 1024), z = q/k/v.
// D(16x16) = emb(16x128) @ W(128x16), K-loop of 32 x wmma_f32_16x16x4_f32.
// Output scattered into [patch][head][p][c] so kernel 2 reads are coalesced.
// ---------------------------------------------------------------------------
__global__ __launch_bounds__(32)
void qkv_proj_kernel(const float* __restrict__ emb,
                     const float* __restrict__ Wq,
                     const float* __restrict__ Wk,
                     const float* __restrict__ Wv,
                     float* __restrict__ ws)
{
    const int mt    = blockIdx.x;          // 0..255  (16-row tile of 4096)
    const int nt    = blockIdx.y;          // 0..63   (16-col tile of 1024)
    const int which = blockIdx.z;          // 0:q 1:k 2:v
    const float* W  = (which == 0) ? Wq : ((which == 1) ? Wk : Wv);
    float* dst      = ws + (size_t)which * SEG;

    const int lane = threadIdx.x;
    const int ln   = lane & 15;            // A: M-row / B: N-col
    const int kh   = lane >> 4;            // K half-select (K = 2*kh, 2*kh+1)
    const int m0   = mt * 16;
    const int n0   = nt * 16;

    v8f acc = {};
    for (int k0 = 0; k0 < CCH; k0 += 4) {
        v2f a, b;
        const float* ap = emb + (size_t)(m0 + ln) * CCH + k0 + 2 * kh;
        a.x = ap[0];              // A[m][k0+2kh]
        a.y = ap[1];              // A[m][k0+2kh+1]
        const float* bp = W + (size_t)(k0 + 2 * kh) * CH + n0 + ln;
        b.x = bp[0];              // B[k0+2kh][n]     (coalesced across lanes)
        b.y = bp[CH];             // B[k0+2kh+1][n]
        acc = WMMA_F32(a, b, acc);
    }

    // D element (m,n): m = v + 8*kh (C/D layout), n = n0 + ln.
    // column n maps to (c_idx = n>>3, head = n&7); row r=(patch,p).
    const int n    = n0 + ln;
    const int cidx = n >> 3;
    const int hh   = n & 7;
    #pragma unroll
    for (int v = 0; v < 8; ++v) {
        const int m     = v + 8 * kh;
        const int r     = m0 + m;
        const int patch = r >> 3;
        const int p     = r & 7;
        dst[(((size_t)patch * HEADS + hh) * PTOK + p) * CCH + cidx] = acc[v];
    }
}

// ---------------------------------------------------------------------------
// Kernel 2: fused per-(patch,head) attention.  Grid = 512*8 blocks, 256 thr
// (8 waves).  Wave w owns attention rows c in [16w,16w+16):
//   attn = Q^T K   (128x128, K-dim = P = 8)  -> 8 accumulator tiles / wave
//   InstanceNorm: softmax is shift-invariant, so only rs=rsqrt(var+eps) needed
//   row softmax fully in registers (width-16 shfl_xor row reductions)
//   bounce own 16 rows through padded LDS to feed ctx = attn @ V^T (128x8)
// Dynamic LDS: 128 x 132 f32 bounce + 16 f32 reduction slots (~66 KB; WGP has
// 320 KB so 4 blocks/WGP still fit).
// ---------------------------------------------------------------------------
__global__ __launch_bounds__(256)
void attn_head_kernel(float* __restrict__ ws)
{
    extern __shared__ float smem[];
    float* attnS = smem;                    // [128][132] padded (bank-safe)
    float* red   = smem + 128 * 132;        // [16] cross-wave sum/sumsq

    const float* Qw  = ws;
    const float* Kw  = ws + SEG;
    const float* Vw  = ws + 2 * SEG;
    float*       Ctx = ws + 3 * SEG;

    const int patch = blockIdx.x >> 3;
    const int h     = blockIdx.x & 7;
    const int tid   = threadIdx.x;
    const int w     = tid >> 5;             // wave 0..7
    const int lane  = tid & 31;
    const int ln    = lane & 15;
    const int kh    = lane >> 4;
    const int c0    = w * 16;               // this wave's c-row block
    const size_t base = ((size_t)patch * HEADS + h) * PTOK * CCH; // [p=0][c=0]

    // ---- attn = Q^T K : A_op[m][k] = Q[p=k][c0+m], B_op[k][n] = K[p=k][e0+n]
    v2f a0, a1;
    {
        const float* q = Qw + base + (size_t)(2 * kh) * CCH + c0 + ln;
        a0.x = q[0];          a0.y = q[CCH];          // p = 2kh, 2kh+1
        a1.x = q[4 * CCH];    a1.y = q[5 * CCH];      // p = 4+2kh, 5+2kh
    }
    v8f acc[8];
    #pragma unroll
    for (int nt = 0; nt < 8; ++nt)
        #pragma unroll
        for (int v = 0; v < 8; ++v) acc[nt][v] = 0.0f;

    #pragma unroll
    for (int nt = 0; nt < 8; ++nt) {
        const float* kp = Kw + base + (size_t)(2 * kh) * CCH + nt * 16 + ln;
        v2f b0, b1;
        b0.x = kp[0];         b0.y = kp[CCH];
        b1.x = kp[4 * CCH];   b1.y = kp[5 * CCH];
        acc[nt] = WMMA_F32(a0, b0, acc[nt]);
        acc[nt] = WMMA_F32(a1, b1, acc[nt]);
    }

    // ---- InstanceNorm statistics over the whole 128x128 map
    float s = 0.0f, ss = 0.0f;
    #pragma unroll
    for (int nt = 0; nt < 8; ++nt)
        #pragma unroll
        for (int v = 0; v < 8; ++v) { const float x = acc[nt][v]; s += x; ss += x * x; }
    #pragma unroll
    for (int off = 16; off >= 1; off >>= 1) {
        s  += __shfl_xor(s,  off, 32);
        ss += __shfl_xor(ss, off, 32);
    }
    if (lane == 0) { red[w] = s; red[8 + w] = ss; }
    __syncthreads();
    float S = 0.0f, SS = 0.0f;
    #pragma unroll
    for (int i = 0; i < 8; ++i) { S += red[i]; SS += red[8 + i]; }
    const float inv16k = 1.0f / 16384.0f;
    const float mu  = S * inv16k;
    const float var = SS * inv16k - mu * mu;
    const float rs  = rsqrtf(var + 1e-5f);   // mean shift cancels in softmax

    // ---- row softmax of rs*attn, fully in registers.
    // C/D layout: slot v holds row (c0 + v) in lanes 0..15 and (c0 + v + 8)
    // in lanes 16..31; e = 16*nt + ln.  Width-16 shfl reduces one row.
    #pragma unroll
    for (int nt = 0; nt < 8; ++nt)
        #pragma unroll
        for (int v = 0; v < 8; ++v) acc[nt][v] *= rs;

    #pragma unroll
    for (int v = 0; v < 8; ++v) {
        float mx = -3.4e38f;
        #pragma unroll
        for (int nt = 0; nt < 8; ++nt) mx = fmaxf(mx, acc[nt][v]);
        #pragma unroll
        for (int off = 8; off >= 1; off >>= 1) mx = fmaxf(mx, __shfl_xor(mx, off, 16));
        float sum = 0.0f;
        #pragma unroll
        for (int nt = 0; nt < 8; ++nt) {
            const float e = __expf(acc[nt][v] - mx);
            acc[nt][v] = e;
            sum += e;
        }
        #pragma unroll
        for (int off = 8; off >= 1; off >>= 1) sum += __shfl_xor(sum, off, 16);
        const float rinv = 1.0f / sum;
        const int row = c0 + v + 8 * kh;
        #pragma unroll
        for (int nt = 0; nt < 8; ++nt)
            attnS[row * 132 + nt * 16 + ln] = acc[nt][v] * rinv;
    }
    __syncthreads();   // uniform; also orders LDS bounce conservatively

    // ---- ctx(c0..c0+15, p=0..7) = attn @ V^T, K-dim = e = 128
    v8f cacc = {};
    const float* vb = Vw + base;  // [p][c]
    for (int k0 = 0; k0 < CCH; k0 += 4) {
        v2f a, b;
        const float* ap = attnS + (size_t)(c0 + ln) * 132 + k0 + 2 * kh;
        a.x = ap[0];
        a.y = ap[1];
        if (ln < 8) {   // B[k][n] = V[p=n][e=k]; n >= 8 is padding
            const float* bp = vb + (size_t)ln * CCH + k0 + 2 * kh;
            b.x = bp[0];
            b.y = bp[1];
        } else {
            b.x = 0.0f;
            b.y = 0.0f;
        }
        cacc = WMMA_F32(a, b, cacc);
    }

    // store ctx[c,p] -> Ctx[patch][p][c*8+h]
    if (ln < 8) {
        #pragma unroll
        for (int v = 0; v < 8; ++v) {
            const int cc = c0 + v + 8 * kh;
            Ctx[((size_t)patch * PTOK + ln) * CH + cc * HEADS + h] = cacc[v];
        }
    }
}

// ---------------------------------------------------------------------------
// Kernel 3: out = Ctx(4096x1024) @ Wout(1024x128) with token-position remap.
// One wave per 16x16 tile: grid (256, 8), 256 K-steps of wmma.
// ---------------------------------------------------------------------------
__global__ __launch_bounds__(32)
void out_proj_kernel(const float* __restrict__ ws,
                     const float* __restrict__ Wout,
                     float* __restrict__ out)
{
    const float* Ctx = ws + 3 * SEG;
    const int mt = blockIdx.x;             // 0..255
    const int nt = blockIdx.y;             // 0..7
    const int lane = threadIdx.x;
    const int ln = lane & 15;
    const int kh = lane >> 4;
    const int m0 = mt * 16;
    const int n0 = nt * 16;

    v8f acc = {};
    for (int k0 = 0; k0 < CH; k0 += 4) {
        v2f a, b;
        const float* ap = Ctx + (size_t)(m0 + ln) * CH + k0 + 2 * kh;
        a.x = ap[0];
        a.y = ap[1];
        const float* bp = Wout + (size_t)(k0 + 2 * kh) * CCH + n0 + ln;
        b.x = bp[0];
        b.y = bp[CCH];
        acc = WMMA_F32(a, b, acc);
    }

    // row r = (patch, p); remap to '(b, d p1 h p2 w p3)' token order
    const int col = n0 + ln;
    #pragma unroll
    for (int v = 0; v < 8; ++v) {
        const int r     = m0 + v + 8 * kh;
        const int patch = r >> 3;
        const int p     = r & 7;
        const int bb  = patch >> 5;
        const int d   = (patch >> 4) & 1;
        const int hh  = (patch >> 2) & 3;
        const int wwp = patch & 3;
        const int p1  = (p >> 2) & 1;
        const int p2  = (p >> 1) & 1;
        const int p3  = p & 1;
        const int orow = bb * 256 + d * 128 + p1 * 64 + hh * 16 + p2 * 8 + wwp * 2 + p3;
        out[(size_t)orow * CCH + col] = acc[v];
    }
}

// ---------------------------------------------------------------------------
extern "C" void kernel_launch(void* const* d_in, const int* in_sizes, int n_in,
                              void* d_out, int out_size, void* d_ws, size_t ws_size,
                              hipStream_t stream)
{
    const float* emb  = (const float*)d_in[0];
    const float* Wq   = (const float*)d_in[1];
    const float* Wk   = (const float*)d_in[2];
    const float* Wv   = (const float*)d_in[3];
    const float* Wout = (const float*)d_in[4];
    float* ws  = (float*)d_ws;      // needs 4*SEG*4 = 64 MB
    float* out = (float*)d_out;

    const size_t smem = (size_t)(128 * 132 + 16) * sizeof(float);   // ~66 KB
    static bool attr_set = [] {
        hipFuncSetAttribute(reinterpret_cast<const void*>(attn_head_kernel),
                            hipFuncAttributeMaxDynamicSharedMemorySize,
                            (int)((128 * 132 + 16) * sizeof(float)));
        return true;
    }();
    (void)attr_set;

    qkv_proj_kernel<<<dim3(256, 64, 3), 32, 0, stream>>>(emb, Wq, Wk, Wv, ws);
    attn_head_kernel<<<dim3(NP * HEADS), 256, smem, stream>>>(ws);
    out_proj_kernel<<<dim3(256, 8), 32, 0, stream>>>(ws, Wout, out);
}